// w2g_84318797955210
// MI455X (gfx1250) — compile-verified
//
#include <hip/hip_runtime.h>

// Elementwise "weight-to-conductance" kernel for MI455X (gfx1250, wave32).
// Memory-bound streaming problem (~450 MiB traffic, ~34 MFLOP -> 0.07 flop/B):
// HBM roofline floor ~20us at 23.3 TB/s; WMMA inapplicable. Optimization =
// B128 coalesced vmem, TH=NT on single-use streams, CDNA5 async global->LDS
// staging (ASYNCcnt) for the two fat streams, branch-free level select.

typedef float  f4 __attribute__((ext_vector_type(4)));
typedef int    i4 __attribute__((ext_vector_type(4)));
// Pointee type of __builtin_amdgcn_global_load_async_to_lds_b128 (from the
// round-2 diagnostic: "int __attribute__((__vector_size__(4*sizeof(int)))) __device__ *")
typedef int    w2g_v4i __attribute__((__vector_size__(4 * sizeof(int))));

#define W2G_AS1 __attribute__((address_space(1)))   // printed as __device__ by clang/HIP
#define W2G_AS3 __attribute__((address_space(3)))   // printed as __shared__

#if defined(__gfx1250__) &&                                        \
    __has_builtin(__builtin_amdgcn_global_load_async_to_lds_b128) && \
    __has_builtin(__builtin_amdgcn_s_wait_asynccnt)
#define W2G_ASYNC 1
#endif

// LDS chunk stride in floats: 16 data + 4 pad. Lane t's chunk starts at bank
// group (5t mod 16) -> conflict-free ds_load_b128 across each 16-lane half.
#define W2G_LSTRIDE 20

// N = 16*16*512*64 NEW_SHAPE elements; each thread covers 4 consecutive flat
// indices (same c,r,m since 4 | 64), i.e. one f4 along the n axis.
__global__ __launch_bounds__(256) void w2g_stream_kernel(
    const float* __restrict__ inp,    // (16,16,64,64) fixed-point weights (float-encoded ints)
    const float* __restrict__ meanG,  // (N,4) per-level mean conductance
    const float* __restrict__ sigG,   // (N,4) per-level sigma
    const float* __restrict__ eps,    // (N)
    const int*   __restrict__ amask,  // (N) bool passed as int32 0/1
    float*       __restrict__ out,    // (4N): [Gp*am | Gn*am | Gp*~am | Gn*~am]
    int n4)                           // N/4
{
    const int tid = threadIdx.x;
    const int t   = blockIdx.x * blockDim.x + tid;
    if (t >= n4) return;

    const size_t base = (size_t)t * 4;
    const size_t N    = (size_t)n4 * 4;

#ifdef W2G_ASYNC
    // Stage the two fat one-pass streams (mean_G/sig_G: 128B/thread) through
    // the CDNA5 async-to-LDS engine (ASYNCcnt): no load-return VGPRs held,
    // overlaps with the direct loads + integer decode below.
    __shared__ float s_mg[256 * W2G_LSTRIDE];
    __shared__ float s_sg[256 * W2G_LSTRIDE];
    {
        const size_t goff = base * 4;  // element offset into (N,4) arrays
#pragma unroll
        for (int e = 0; e < 4; ++e) {
            __builtin_amdgcn_global_load_async_to_lds_b128(
                (W2G_AS1 w2g_v4i*)(uintptr_t)(const void*)(meanG + goff + (size_t)e * 4),
                (W2G_AS3 w2g_v4i*)(s_mg + tid * W2G_LSTRIDE + e * 4),
                /*offset=*/0, /*cpol=*/0);
            __builtin_amdgcn_global_load_async_to_lds_b128(
                (W2G_AS1 w2g_v4i*)(uintptr_t)(const void*)(sigG + goff + (size_t)e * 4),
                (W2G_AS3 w2g_v4i*)(s_sg + tid * W2G_LSTRIDE + e * 4),
                /*offset=*/0, /*cpol=*/0);
        }
    }
#endif

    // Decode NEW_SHAPE flat index: (((c*16+r)*512 + m)*64 + n)
    const int n  = (int)(base & 63);          // fastest dim (4 consecutive)
    const int m  = (int)((base >> 6) & 511);  // m = s2*8 + b, b MSB-first slice
    const int cr = (int)(base >> 15);         // c*16 + r
    const int s2 = m >> 3;
    const int sh = (7 - (m & 7)) * 2;         // bitslice shift, MSB first

    // Input element (c,r,s2,n): reused by 8 bit-slices -> default (RT) load.
    const size_t in_idx = ((size_t)(cr * 64 + s2)) * 64 + (size_t)n;
    f4 xv = *(const f4*)(inp + in_idx);

    // Single-use streams: non-temporal loads keep them out of L2/WGP$.
    f4 ev = __builtin_nontemporal_load((const f4*)(eps + base));
    i4 mv = __builtin_nontemporal_load((const i4*)(amask + base));

#ifdef W2G_ASYNC
    // All 8 async B128s done -> their LDS writes are visible to this wave.
    __builtin_amdgcn_s_wait_asynccnt(0);
#endif

    f4 o_ap, o_an, o_ip, o_in;
#pragma unroll
    for (int e = 0; e < 4; ++e) {
        // All 4 levels live in one 16B block; the HBM cacheline is fetched
        // regardless of which level each lane needs, so load all 4 and select
        // (v_cndmask) instead of doing a divergent per-code gather.
#ifdef W2G_ASYNC
        f4 mg = *(const f4*)(s_mg + tid * W2G_LSTRIDE + e * 4);  // ds_load_b128
        f4 sg = *(const f4*)(s_sg + tid * W2G_LSTRIDE + e * 4);
#else
        f4 mg = __builtin_nontemporal_load((const f4*)(meanG + (base + (size_t)e) * 4));
        f4 sg = __builtin_nontemporal_load((const f4*)(sigG  + (base + (size_t)e) * 4));
#endif

        const float epsv = ev[e];
        const float g0 = fmaf(epsv, sg.x, mg.x);
        const float g1 = fmaf(epsv, sg.y, mg.y);
        const float g2 = fmaf(epsv, sg.z, mg.z);
        const float g3 = fmaf(epsv, sg.w, mg.w);

        const float x  = xv[e];
        // pos = max(x,0), neg = max(-x,0); values are exact integers <= 32767
        const int ip = (int)fmaxf(x, 0.0f);
        const int im = (int)fmaxf(-x, 0.0f);
        const int cp = (ip >> sh) & 3;
        const int cn = (im >> sh) & 3;

        const float Gp = (cp == 0) ? g0 : (cp == 1) ? g1 : (cp == 2) ? g2 : g3;
        const float Gn = (cn == 0) ? g0 : (cn == 1) ? g1 : (cn == 2) ? g2 : g3;

        const float am  = (float)mv[e];   // 0.0 or 1.0
        const float nam = 1.0f - am;

        o_ap[e] = Gp * am;
        o_an[e] = Gn * am;
        o_ip[e] = Gp * nam;
        o_in[e] = Gn * nam;
    }

    // Four streaming B128 stores, non-temporal (write-once, never re-read).
    __builtin_nontemporal_store(o_ap, (f4*)(out + base));
    __builtin_nontemporal_store(o_an, (f4*)(out + N + base));
    __builtin_nontemporal_store(o_ip, (f4*)(out + 2 * N + base));
    __builtin_nontemporal_store(o_in, (f4*)(out + 3 * N + base));
}

extern "C" void kernel_launch(void* const* d_in, const int* in_sizes, int n_in,
                              void* d_out, int out_size, void* d_ws, size_t ws_size,
                              hipStream_t stream) {
    // setup_inputs() order: input, mean_G, sig_G, eps, active_mask
    const float* inp  = (const float*)d_in[0];
    const float* mG   = (const float*)d_in[1];
    const float* sG   = (const float*)d_in[2];
    const float* eps  = (const float*)d_in[3];
    const int*   am   = (const int*)d_in[4];   // bool -> int32 per harness contract
    float*       out  = (float*)d_out;

    const int N  = in_sizes[3];   // prod(NEW_SHAPE) = 8,388,608
    const int n4 = N / 4;

    dim3 block(256);              // 8 wave32 waves per block
    dim3 grid((n4 + 255) / 256);  // 8192 blocks
    hipLaunchKernelGGL(w2g_stream_kernel, grid, block, 0, stream,
                       inp, mG, sG, eps, am, out, n4);
}